// SlayerNet_2199023255635
// MI455X (gfx1250) — compile-verified
//
#include <hip/hip_runtime.h>
#include <math.h>

#define NUM_T   100
#define TAU     10.0f
#define THETA   1.0f
#define EPS_K   0.01f
#define PAD     101        // lw stride: odd -> LDS bank-conflict-free column reads
#define PPAD    113        // psp stride: odd, >= 112 so unmasked D-writes are safe
#define ESZ     212        // reversed band: srm_rev[i] = band(111 - i)
#define ROWS    64         // samples per block
#define THREADS 128        // 4 waves (wave32)
#define NWAVES  (THREADS / 32)
#define NRT     (ROWS / 16)          // 4 row tiles
#define NCT     ((NUM_T + 15) / 16)  // 7 col tiles (cols 100..111 computed, never read)

// Manual LDS layout inside ONE shared aggregate (base 0): srm_rev at offset 0
// so B-fragment offsets (<=211 dwords) fit the 8-bit DS immediate fields.
#define OFF_SRM  0
#define OFF_REFK (OFF_SRM + ESZ)          // 212
#define OFF_WLDS (OFF_REFK + NUM_T)       // 312
#define OFF_KBRK (OFF_WLDS + 32)          // 344 (2 ints stored as float bits)
#define OFF_LW   (OFF_KBRK + 2)           // 346
#define OFF_PSP  (OFF_LW + ROWS * PAD)    // 346 + 6464 = 6810
#define SMEM_TOT (OFF_PSP + ROWS * PPAD)  // 6810 + 7232 = 14042 floats (~56.2 KB)

typedef float v2f __attribute__((ext_vector_type(2)));
typedef float v8f __attribute__((ext_vector_type(8)));

__device__ __forceinline__ float alpha_val(float t) {
    return (t / TAU) * expf(1.0f - t / TAU);
}

__global__ __launch_bounds__(THREADS) void slayer_fused_kernel(
    const float* __restrict__ x,   // (N, 2,4,4, 100) -> (N, 32, 100)
    const float* __restrict__ Wp,  // (1, 2,4,4)      -> (32)
    float* __restrict__ out,       // (N, 1,1,1, 100) -> (N, 100)
    int N)
{
    __shared__ float smem[SMEM_TOT];
    float* const srm_rev = smem + OFF_SRM;   // reversed zero-padded Toeplitz band
    float* const refk    = smem + OFF_REFK;  // refractory kernel (zero-padded)
    float* const wlds    = smem + OFF_WLDS;
    float* const kbrk    = smem + OFF_KBRK;  // break indices (float-bitcast ints)
    float* const lw      = smem + OFF_LW;    // weighted input; reused for spikes
    float* const psp     = smem + OFF_PSP;   // post-SRM membrane potential

    const int tid  = threadIdx.x;
    const int lane = tid & 31;
    const int wave = tid >> 5;
    const int n0   = blockIdx.x * ROWS;

    // ---- phase 0: weights; break indices (python: first t>tau with |v|<eps) ----
    if (tid < 32) wlds[tid] = Wp[tid];
    if (tid == 0) {
        int ks = NUM_T, kr = NUM_T;
        for (int t = 11; t < NUM_T; ++t) {
            float v = alpha_val((float)t);           // srm: mult=+1
            if (fabsf(v) < EPS_K) { ks = t; break; }
        }
        for (int t = 11; t < NUM_T; ++t) {
            float v = -2.0f * alpha_val((float)t);   // ref: mult=-2*THETA*SCALE_REF
            if (fabsf(v) < EPS_K) { kr = t; break; }
        }
        kbrk[0] = __int_as_float(ks);
        kbrk[1] = __int_as_float(kr);
    }
    __syncthreads();

    // ---- phase 1a: fill reversed band + refractory kernel ----
    // srm_rev[i] = SRM band value at displacement d = 111 - i (zero outside [0,ks))
    {
        const int ks = __float_as_int(kbrk[0]);
        const int kr = __float_as_int(kbrk[1]);
        for (int i = tid; i < ESZ; i += THREADS) {
            int d = 111 - i;
            srm_rev[i] = (d >= 0 && d < ks) ? alpha_val((float)d) : 0.0f;
        }
        for (int i = tid; i < NUM_T; i += THREADS)
            refk[i] = (i < kr) ? -2.0f * alpha_val((float)i) : 0.0f;
    }

    // ---- phase 1b: weighted[n,t] = dot32(x[n,:,t], W), float4 over t ----
    const int TASKS = ROWS * (NUM_T / 4);   // 64 * 25
    for (int idx = tid; idx < TASKS; idx += THREADS) {
        int nl = idx / (NUM_T / 4);
        int q  = idx - nl * (NUM_T / 4);
        int n  = n0 + nl;
        if (n < N) {
            size_t base = (size_t)n * (32 * NUM_T) + 4 * q;  // 16B-aligned
            float s0 = 0.f, s1 = 0.f, s2 = 0.f, s3 = 0.f;
            #pragma unroll
            for (int c = 0; c < 32; ++c) {
                float4 v = *(const float4*)(x + base + (size_t)c * NUM_T);
                float w = wlds[c];
                s0 += v.x * w; s1 += v.y * w; s2 += v.z * w; s3 += v.w * w;
            }
            float* dst = &lw[nl * PAD + 4 * q];
            dst[0] = s0; dst[1] = s1; dst[2] = s2; dst[3] = s3;
        }
    }
    __syncthreads();

    // ---- phase 2: psp = weighted x SRM_MAT via V_WMMA_F32_16X16X4_F32 ----
    // Base dword indices are pinned into VGPRs so the large constant parts fold
    // into the base register once, leaving k (<=97 / <=211 dwords) as the
    // 8-bit DS immediate offsets: ds_load_2addr_b32 base offset0:k offset1:k+1.
    {
        const int m    = lane & 15;          // row / col within 16x16 tile
        const int koff = (lane >> 4) * 2;    // K pair held by this half-wave
        for (int tile = wave; tile < NRT * NCT; tile += NWAVES) {
            int rt = tile / NCT, ct = tile - rt * NCT;
            int aidx = OFF_LW + (rt * 16 + m) * PAD + koff;   // A base (dwords)
            int bidx = 111 - (ct * 16 + m) + koff;            // B base (OFF_SRM=0)
            asm("" : "+v"(aidx), "+v"(bidx));                 // pin full bases
            const float* arow = smem + aidx;
            const float* brow = smem + bidx;
            v8f acc = {0.f, 0.f, 0.f, 0.f, 0.f, 0.f, 0.f, 0.f};
            #pragma unroll
            for (int k = 0; k < NUM_T; k += 4) {
                v2f a, b;
                a.x = arow[k];            // A[m][k+koff]
                a.y = arow[k + 1];        // A[m][k+koff+1]
                b.x = brow[k];            // SRM_MAT[k+koff    ][u]
                b.y = brow[k + 1];        // SRM_MAT[k+koff + 1][u]
                acc = __builtin_amdgcn_wmma_f32_16x16x4_f32(
                    false, a, false, b, (short)0, acc, false, false);
            }
            // D write, unmasked: cols 100..111 land in psp padding, never read
            int rbase = rt * 16 + (lane >> 4) * 8;
            int u     = ct * 16 + m;
            #pragma unroll
            for (int r = 0; r < 8; ++r)
                psp[(rbase + r) * PPAD + u] = acc[r];   // * TS (=1.0)
        }
    }
    __syncthreads();

    // ---- phase 3: sequential spike scan (parallel over rows) ----
    if (tid < ROWS) {
        float* prow = &psp[tid * PPAD];
        float* srow = &lw[tid * PAD];     // lw reused as spike staging
        const int kr = __float_as_int(kbrk[1]);
        for (int t = 0; t < NUM_T; ++t) {
            float u = prow[t];
            bool fire = (u >= THETA);
            srow[t] = fire ? 1.0f : 0.0f;  // s / TS, TS = 1
            if (fire) {
                int jmax = NUM_T - t;
                if (kr < jmax) jmax = kr;  // refk[j]=0 beyond kr anyway
                for (int j = 1; j < jmax; ++j)
                    prow[t + j] += refk[j];
            }
        }
    }
    __syncthreads();

    // ---- phase 4: coalesced write-out ----
    for (int idx = tid; idx < ROWS * NUM_T; idx += THREADS) {
        int nl = idx / NUM_T;
        int t  = idx - nl * NUM_T;
        int n  = n0 + nl;
        if (n < N) out[(size_t)n * NUM_T + t] = lw[nl * PAD + t];
    }
}

extern "C" void kernel_launch(void* const* d_in, const int* in_sizes, int n_in,
                              void* d_out, int out_size, void* d_ws, size_t ws_size,
                              hipStream_t stream) {
    const float* x  = (const float*)d_in[0];   // (16384,2,4,4,100) f32
    const float* Wp = (const float*)d_in[1];   // (1,2,4,4) f32
    float* out = (float*)d_out;                // (16384,1,1,1,100) f32

    int N = in_sizes[0] / (32 * NUM_T);        // 16384
    int blocks = (N + ROWS - 1) / ROWS;        // 256
    slayer_fused_kernel<<<blocks, THREADS, 0, stream>>>(x, Wp, out, N);
}